// DeHoog_36249523978646
// MI455X (gfx1250) — compile-verified
//
#include <hip/hip_runtime.h>

// De Hoog inverse Laplace transform for MI455X (gfx1250).
// - No matrix product exists in this op -> WMMA intentionally not used.
// - CDNA5 paths used: global_load_async_to_lds_b128 (ASYNCcnt) staging +
//   s_wait_asynccnt; complex math written on 2-wide float vectors in
//   v_pk_fma_f32 / v_pk_add_f32-friendly form (VOP3P packed fp32) to cut
//   VALU instruction count (kernel is VALU-bound per roofline).
// - QD recurrence fully unrolled via template recursion so all array indices
//   are compile-time constants -> q/e arrays live in VGPRs (no scratch).

#define NTERMS 33
#define MM 16
#define TPB 128

typedef float v2f __attribute__((ext_vector_type(2)));

// complex mul: (ax bx - ay by, ax by + ay bx)
//   = a * (bx,bx) + (-ay, ax) * (by,by)     -> v_pk_mul + v_pk_fma
__device__ __forceinline__ v2f cmul(v2f a, v2f b) {
    v2f m = { -a.y, a.x };
    return a * b.xx + m * b.yy;
}
// complex div: a * conj(b) / |b|^2
//   num = a * (bx,bx) + (ay, -ax) * (by,by) = (ax bx + ay by, ay bx - ax by)
__device__ __forceinline__ v2f cdiv(v2f a, v2f b) {
    float s = 1.0f / (b.x * b.x + b.y * b.y);
    v2f m = { a.y, -a.x };
    v2f num = a * b.xx + m * b.yy;
    v2f sv = { s, s };
    return num * sv;
}
// Continued fraction step: A_n = A_{n-1} + d_n*z*A_{n-2} (and same for B).
__device__ __forceinline__ void cf_step(v2f d, v2f z, v2f& Ap, v2f& Ac, v2f& Bp, v2f& Bc) {
    v2f dz = cmul(d, z);
    v2f An = Ac + cmul(dz, Ap);
    v2f Bn = Bc + cmul(dz, Bp);
    Ap = Ac; Ac = An; Bp = Bc; Bc = Bn;
}

// One QD iteration r=R (source-level unroll so inner bounds are constexpr).
template<int R>
struct QDIter {
    static __device__ __forceinline__ void run(v2f (&q)[32], v2f (&e)[NTERMS], v2f z,
            v2f& Ap, v2f& Ac, v2f& Bp, v2f& Bc, v2f& d31, v2f& d32) {
        constexpr int Le = 2 * (MM - R) + 1;
        // e_r[j] = q[j+1] - q[j] + e_{r-1}[j+1]   (in-place, ascending j is safe)
        #pragma unroll
        for (int j = 0; j < Le; ++j) {
            e[j] = (q[j + 1] - q[j]) + e[j + 1];   // 2x v_pk_add_f32 (neg mod)
        }
        v2f d_even = -e[0];                        // d_{2R}
        cf_step(d_even, z, Ap, Ac, Bp, Bc);
        if constexpr (R == MM) {
            d32 = d_even;
        } else {
            constexpr int Lq = 2 * (MM - R);
            // q_{r+1}[j] = q[j+1] * e[j+1] / e[j]  (in-place, ascending j is safe)
            #pragma unroll
            for (int j = 0; j < Lq; ++j) {
                q[j] = cdiv(cmul(q[j + 1], e[j + 1]), e[j]);
            }
            v2f d_odd = -q[0];                     // d_{2R+1}
            cf_step(d_odd, z, Ap, Ac, Bp, Bc);
            if constexpr (R == MM - 1) d31 = d_odd;
            QDIter<R + 1>::run(q, e, z, Ap, Ac, Bp, Bc, d31, d32);
        }
    }
};

__global__ __launch_bounds__(TPB, 1) void dehoog_kernel(
        const float* __restrict__ fpr, const float* __restrict__ fpi,
        const float* __restrict__ tiv, const float* __restrict__ Tv,
        float* __restrict__ out, int S, int D, int npoints)
{
    __shared__ __align__(16) float tile[2 * TPB * NTERMS];  // [re rows | im rows]
    const int tid  = threadIdx.x;
    const int base = blockIdx.x * TPB;

    // Touch LDS from IR once (keeps optimizer honest about LDS liveness);
    // ordered before the async DMA writes by the barrier below.
    tile[tid] = 0.0f;
    tile[TPB * NTERMS + tid] = 0.0f;
    __syncthreads();

    // ---- Async global -> LDS staging (CDNA5 path, tracked by ASYNCcnt) ----
    constexpr int TILE_BYTES = TPB * NTERMS * 4;      // 16896 (16B-aligned per block)
    constexpr int CHUNKS     = TILE_BYTES / 16;       // 1056 b128 chunks per array
    unsigned ldsRe = (unsigned)(unsigned long long)&tile[0];
    unsigned ldsIm = ldsRe + (unsigned)TILE_BYTES;
    unsigned long long gRe = (unsigned long long)(fpr + (size_t)base * NTERMS);
    unsigned long long gIm = (unsigned long long)(fpi + (size_t)base * NTERMS);
    // defensive cap for a (non-occurring here) partial last tile
    long long availB = ((long long)npoints - (long long)base) * NTERMS * 4;
    int chunksAvail = CHUNKS;
    if (availB < (long long)TILE_BYTES) chunksAvail = (int)(availB / 16);

    #pragma unroll
    for (int k = 0; k < (CHUNKS + TPB - 1) / TPB; ++k) {
        int c = tid + k * TPB;
        if (c < chunksAvail) {
            unsigned voff = (unsigned)c * 16u;
            asm volatile("global_load_async_to_lds_b128 %0, %1, %2"
                         :: "v"(ldsRe + voff), "v"(voff), "s"(gRe) : "memory");
            asm volatile("global_load_async_to_lds_b128 %0, %1, %2"
                         :: "v"(ldsIm + voff), "v"(voff), "s"(gIm) : "memory");
        }
    }
    asm volatile("s_wait_asynccnt 0" ::: "memory");
    __syncthreads();

    const int n = base + tid;
    if (n >= npoints) return;

    const int rowR = tid * NTERMS;
    const int rowI = TPB * NTERMS + rowR;

    // Per-time-point contour parameters (ti/T are tiny -> cached global loads).
    const int   sidx  = (n / D) % S;
    const float Ts    = Tv[sidx];
    const float tis   = tiv[sidx];
    const float Tsc   = 2.0f * Ts;                                   // SCALE*T
    const float gamma = 1.0e-3f + 4.605170185988091f / (2.0f * Tsc); // ALPHA - ln(TOL)/(SCALE*Tsc)
    const float ang   = 3.14159265358979323846f * tis / Tsc;
    const v2f z = { cosf(ang), sinf(ang) };                          // exp(i*pi*ti/Tsc)

    // a_0 = fp[0]/2; a_k = fp[k].  Stream a[] out of LDS while building q.
    v2f ap = { 0.5f * tile[rowR], 0.5f * tile[rowI] };
    const v2f d0 = ap;
    v2f q[32];
    #pragma unroll
    for (int j = 0; j < 32; ++j) {
        v2f an = { tile[rowR + j + 1], tile[rowI + j + 1] };
        q[j] = cdiv(an, ap);                       // q1[j] = a[j+1]/a[j]
        ap = an;
    }
    v2f e[NTERMS];
    #pragma unroll
    for (int k = 0; k < NTERMS; ++k) { e[k] = (v2f){ 0.0f, 0.0f }; }

    // Continued fraction scan state; consume d_n in order as QD produces them.
    v2f Ap = { 0.0f, 0.0f }, Ac = d0;
    v2f Bp = { 1.0f, 0.0f }, Bc = { 1.0f, 0.0f };
    v2f d31 = { 0.0f, 0.0f }, d32 = { 0.0f, 0.0f };
    cf_step(-q[0], z, Ap, Ac, Bp, Bc);             // d_1 = -q1[0]
    QDIter<1>::run(q, e, z, Ap, Ac, Bp, Bc, d31, d32);

    // Double acceleration remainder: brem = (1 + (d_{2M-1}-d_{2M}) z)/2,
    // rem = -brem (1 - sqrt(1 + d_{2M} z / brem^2))   (principal sqrt).
    v2f w    = cmul(d31 - d32, z);
    v2f brem = { 0.5f * (1.0f + w.x), 0.5f * w.y };
    v2f h    = cdiv(cmul(d32, z), cmul(brem, brem));
    v2f u    = { 1.0f + h.x, h.y };
    float mag = sqrtf(u.x * u.x + u.y * u.y);
    float wr  = sqrtf(fmaxf(0.5f * (mag + u.x), 0.0f));
    float wi  = copysignf(sqrtf(fmaxf(0.5f * (mag - u.x), 0.0f)), u.y);
    v2f rem = cmul(-brem, (v2f){ 1.0f - wr, -wi });

    v2f Af = Ac + cmul(rem, Ap);
    v2f Bf = Bc + cmul(rem, Bp);
    float den = Bf.x * Bf.x + Bf.y * Bf.y;
    float val = (Af.x * Bf.x + Af.y * Bf.y) / den;      // Re(Af/Bf)
    out[n] = (expf(gamma * tis) / Tsc) * val;
}

extern "C" void kernel_launch(void* const* d_in, const int* in_sizes, int n_in,
                              void* d_out, int out_size, void* d_ws, size_t ws_size,
                              hipStream_t stream) {
    const float* fpr = (const float*)d_in[0];
    const float* fpi = (const float*)d_in[1];
    const float* tiv = (const float*)d_in[2];
    const float* Tv  = (const float*)d_in[3];
    float* out = (float*)d_out;
    const int S = in_sizes[2];          // 512
    const int D = 32;                   // feature dim from reference setup
    const int npoints = out_size;       // B*S*D = 524288
    const int blocks = (npoints + TPB - 1) / TPB;
    dehoog_kernel<<<blocks, TPB, 0, stream>>>(fpr, fpi, tiv, Tv, out, S, D, npoints);
}